// Model_62423054680326
// MI455X (gfx1250) — compile-verified
//
#include <hip/hip_runtime.h>

// Problem constants (from the reference file)
constexpr int Bc = 32, Nc = 512, Gc = 64, Tc = 16, Dc = 256;
constexpr int TD    = Tc * Dc;     // 4096 floats between consecutive "type" rows
constexpr int SROWS = Bc * Nc;     // 16384 scope type rows
constexpr int GROWS = Bc * Gc;     // 2048 goal type rows
constexpr float NEGV = -1.0e10f;

typedef __attribute__((ext_vector_type(2))) float v2f;
typedef __attribute__((ext_vector_type(4))) float v4f;
typedef __attribute__((ext_vector_type(8))) float v8f;

// ---------------------------------------------------------------------------
// Kernel 0: pack the strided type rows (stride 16KB) into a dense 16MB buffer
// so the 500k row-gathers touch fully-used, L2-resident cachelines.
// ---------------------------------------------------------------------------
__global__ void densify_scope_type(const float* __restrict__ scope,
                                   float* __restrict__ dst) {
    const int row = blockIdx.x;      // 0..SROWS-1
    const int t   = threadIdx.x;     // 0..Dc-1
    dst[(size_t)row * Dc + t] = scope[(size_t)row * TD + t];
}

// ---------------------------------------------------------------------------
// Kernel 1: Y = goal_type (2048x256) @ W^T (256x256) via V_WMMA_F32_16X16X4_F32
// One wave per 16x16 output tile, K stepped by 4.
//   A 16x4 f32 layout : lanes 0-15 -> rows, {v0,v1} = K 0,1 ; lanes 16-31 = K 2,3
//   B 4x16 f32 layout : mirrored (lanes = cols), B[k][j] = W[d0+j][k]
//   C/D 16x16 layout  : VGPR v holds row v (lanes 0-15) / row v+8 (lanes 16-31)
// ---------------------------------------------------------------------------
__global__ void goalW_gemm(const float* __restrict__ goal,   // (B,G,T,D) flat
                           const float* __restrict__ W,      // (D,D) row-major
                           float* __restrict__ Y) {          // (GROWS, Dc)
    const int wave   = (blockIdx.x * blockDim.x + threadIdx.x) >> 5;
    const int lane   = threadIdx.x & 31;
    const int tile_d = wave & 15;          // 256/16 = 16 column tiles
    const int tile_g = wave >> 4;          // 2048/16 = 128 row tiles
    const int g0 = tile_g * 16, d0 = tile_d * 16;
    const int half = lane >> 4;            // 0 -> K 0..1, 1 -> K 2..3
    const int rc   = lane & 15;            // A-row / B-col for this lane

    // goal_type row g lives at flat offset g*T*D (the [.., .., 0, :] slice)
    const float* aRow = goal + (size_t)(g0 + rc) * TD + half * 2;
    const float* bRow = W    + (size_t)(d0 + rc) * Dc + half * 2;

    v8f acc = {};
#pragma unroll 4
    for (int k = 0; k < Dc; k += 4) {
        v2f a = *(const v2f*)(aRow + k);
        v2f b = *(const v2f*)(bRow + k);
        acc = __builtin_amdgcn_wmma_f32_16x16x4_f32(
            /*neg_a=*/false, a, /*neg_b=*/false, b,
            /*c_mod=*/(short)0, acc, /*reuse_a=*/false, /*reuse_b=*/false);
    }
#pragma unroll
    for (int v = 0; v < 8; ++v)
        Y[(size_t)(g0 + v + 8 * half) * Dc + d0 + rc] = acc[v];
}

// ---------------------------------------------------------------------------
// Kernel 2: lemma_preds[e] = dot(scope_type[src[e]], Y[tgt[e]]) + bias
// One wave per edge; each lane owns 8 contiguous floats; wave32 shfl reduce.
// All gathered rows are L2-resident (16MB + 2MB working set vs 192MB L2).
// ---------------------------------------------------------------------------
__global__ void edge_bilinear(const float* __restrict__ scopeT, int srowStride,
                              const float* __restrict__ Y,
                              const int* __restrict__ edge,   // (2,E) flat
                              const float* __restrict__ bias,
                              float* __restrict__ out, int E) {
    const int e    = (blockIdx.x * blockDim.x + threadIdx.x) >> 5;
    const int lane = threadIdx.x & 31;
    if (e >= E) return;
    const int s = edge[e];
    const int t = edge[E + e];
    const float* sp = scopeT + (size_t)s * srowStride + lane * 8;
    const float* yp = Y      + (size_t)t * Dc        + lane * 8;
    v4f s0 = *(const v4f*)sp, s1 = *(const v4f*)(sp + 4);
    v4f y0 = *(const v4f*)yp, y1 = *(const v4f*)(yp + 4);
    float acc = s0.x*y0.x + s0.y*y0.y + s0.z*y0.z + s0.w*y0.w
              + s1.x*y1.x + s1.y*y1.y + s1.z*y1.z + s1.w*y1.w;
#pragma unroll
    for (int off = 16; off >= 1; off >>= 1)
        acc += __shfl_xor(acc, off, 32);
    if (lane == 0) out[e] = acc + bias[0];
}

// ---------------------------------------------------------------------------
// Kernel 3: lm_preds[m,n] = dot(mask_reprs[m], type_reprs[batch_pts[m], n])
// One block (8 waves) per m; mask row cached in registers; wave per column n.
// ---------------------------------------------------------------------------
__global__ void lm_scores(const float* __restrict__ scope,    // full tensor
                          const float* __restrict__ scopeT, int srowStride,
                          const int* __restrict__ lm_idx,
                          const int* __restrict__ bpts,
                          const unsigned char* __restrict__ tree_mask, // (B,N) bool
                          float* __restrict__ out) {          // (M,N)
    const int m    = blockIdx.x;
    const int wave = threadIdx.x >> 5;   // 0..7
    const int lane = threadIdx.x & 31;
    const int r = lm_idx[m];             // token row in reshape(-1, D)
    const int b = bpts[m];

    const float* mp = scope + (size_t)r * Dc + lane * 8;
    v4f m0 = *(const v4f*)mp, m1 = *(const v4f*)(mp + 4);

    for (int n = wave; n < Nc; n += 8) {
        const float* cp = scopeT + (size_t)(b * Nc + n) * srowStride + lane * 8;
        v4f c0 = *(const v4f*)cp, c1 = *(const v4f*)(cp + 4);
        float acc = m0.x*c0.x + m0.y*c0.y + m0.z*c0.z + m0.w*c0.w
                  + m1.x*c1.x + m1.y*c1.y + m1.z*c1.z + m1.w*c1.w;
#pragma unroll
        for (int off = 16; off >= 1; off >>= 1)
            acc += __shfl_xor(acc, off, 32);
        if (lane == 0)
            out[(size_t)m * Nc + n] = tree_mask[b * Nc + n] ? acc : NEGV;
    }
}

// ---------------------------------------------------------------------------
extern "C" void kernel_launch(void* const* d_in, const int* in_sizes, int n_in,
                              void* d_out, int out_size, void* d_ws, size_t ws_size,
                              hipStream_t stream) {
    const float* scope = (const float*)d_in[0];           // (B,N,T,D) f32
    const float* goal  = (const float*)d_in[1];           // (B,G,T,D) f32
    const float* W     = (const float*)d_in[2];           // (1,D,D)   f32
    const float* bias  = (const float*)d_in[3];           // (1,)      f32
    const int*   edge  = (const int*)d_in[4];             // (2,E)     i32
    const int*   lmidx = (const int*)d_in[5];             // (M,)      i32
    const int*   bpts  = (const int*)d_in[6];             // (M,)      i32
    const unsigned char* tmask = (const unsigned char*)d_in[7]; // (B,N) bool

    const int E = in_sizes[4] / 2;
    const int M = in_sizes[5];

    // Workspace layout: [ Y : 2048*256 f32 = 2MB ][ dense type rows : 16MB ]
    float*  Y          = (float*)d_ws;
    size_t  yBytes     = (size_t)GROWS * Dc * sizeof(float);
    float*  dense      = (float*)((char*)d_ws + yBytes);
    size_t  denseBytes = (size_t)SROWS * Dc * sizeof(float);

    const float* scopeT = scope;
    int stride = TD;                                      // strided fallback
    if (ws_size >= yBytes + denseBytes) {
        densify_scope_type<<<SROWS, Dc, 0, stream>>>(scope, dense);
        scopeT = dense;
        stride = Dc;
    }

    // Y = goal_type @ W^T  (WMMA f32 16x16x4): 2048 tiles, 8 waves/block
    goalW_gemm<<<(GROWS / 16) * (Dc / 16) / 8, 256, 0, stream>>>(goal, W, Y);

    // lemma_preds: one wave per edge, 8 edges per 256-thread block
    edge_bilinear<<<(E + 7) / 8, 256, 0, stream>>>(scopeT, stride, Y, edge, bias,
                                                   (float*)d_out, E);

    // lm_preds at offset E in d_out
    lm_scores<<<M, 256, 0, stream>>>(scope, scopeT, stride, lmidx, bpts, tmask,
                                     (float*)d_out + E);
}